// GCNEncoder_4913442587254
// MI455X (gfx1250) — compile-verified
//
#include <hip/hip_runtime.h>
#include <hip/hip_bf16.h>

typedef __attribute__((ext_vector_type(2))) float v2f;
typedef __attribute__((ext_vector_type(8))) float v8f;

#define NNODES 50000
#define NEDGES 800000
#define DH 128
#define DOUT 64

// ---------------------------------------------------------------- utilities
__global__ __launch_bounds__(256) void zero_f32(float* __restrict__ p, int n) {
    int i = blockIdx.x * blockDim.x + threadIdx.x;
    if (i < n) p[i] = 0.0f;
}

__global__ __launch_bounds__(256) void deg_kernel(const int* __restrict__ dst,
                                                  float* __restrict__ deg, int e) {
    int i = blockIdx.x * blockDim.x + threadIdx.x;
    if (i < e) atomicAdd(&deg[dst[i]], 1.0f);
}

__global__ __launch_bounds__(256) void dinv_kernel(const float* __restrict__ deg,
                                                   float* __restrict__ dinv, int n) {
    int i = blockIdx.x * blockDim.x + threadIdx.x;
    if (i < n) dinv[i] = rsqrtf(deg[i] + 1.0f);
}

__global__ __launch_bounds__(256) void ecoef_kernel(const int* __restrict__ src,
                                                    const int* __restrict__ dst,
                                                    const float* __restrict__ dinv,
                                                    float* __restrict__ ec, int e) {
    int i = blockIdx.x * blockDim.x + threadIdx.x;
    if (i < e) ec[i] = dinv[src[i]] * dinv[dst[i]];
}

// ------------------------------------------------- GEMM: T[N,128] = H[N,128] @ W[128,128]
// Block = 256 threads = 8 waves. Block owns a 16-row strip; wave w owns column tile w.
// fp32 WMMA 16x16x4, K=128 -> 32 wmma ops per wave.
__global__ __launch_bounds__(256) void gemm_h(const float* __restrict__ H,
                                              const float* __restrict__ W,
                                              float* __restrict__ T) {
    __shared__ float As[16 * DH];
    const int row0 = blockIdx.x * 16;
    // stage A strip (16x128 = 8KB) into LDS, coalesced
    for (int i = threadIdx.x; i < 16 * DH; i += 256)
        As[i] = H[row0 * DH + i];
    __syncthreads();

    const int wave = threadIdx.x >> 5;
    const int lane = threadIdx.x & 31;
    const int col0 = wave * 16;
    const int m    = lane & 15;           // row (A) / col (B,C)
    const int kp   = (lane >> 4) << 1;    // K sub-pair: lanes 0-15 -> K=0,1; 16-31 -> K=2,3

    v8f acc = {};
    #pragma unroll 4
    for (int k = 0; k < DH; k += 4) {
        v2f a, b;
        a.x = As[m * DH + k + kp];
        a.y = As[m * DH + k + kp + 1];
        b.x = W[(k + kp)     * DH + col0 + m];
        b.y = W[(k + kp + 1) * DH + col0 + m];
        acc = __builtin_amdgcn_wmma_f32_16x16x4_f32(false, a, false, b,
                                                    (short)0, acc, false, false);
    }
    // C layout: VGPR r -> row r (lanes 0-15) / row r+8 (lanes 16-31), col = lane&15
    const int rbase = row0 + ((lane >> 4) << 3);
    #pragma unroll
    for (int r = 0; r < 8; ++r)
        T[(rbase + r) * DH + col0 + m] = acc[r];
}

// ----------------------------------- GEMM: OUT[N,64] = H[N,128] @ Wout[128,64] + bout
__global__ __launch_bounds__(128) void gemm_o(const float* __restrict__ H,
                                              const float* __restrict__ W,
                                              const float* __restrict__ bias,
                                              float* __restrict__ OUT) {
    __shared__ float As[16 * DH];
    const int row0 = blockIdx.x * 16;
    for (int i = threadIdx.x; i < 16 * DH; i += 128)
        As[i] = H[row0 * DH + i];
    __syncthreads();

    const int wave = threadIdx.x >> 5;    // 0..3 (64/16 column tiles)
    const int lane = threadIdx.x & 31;
    const int col0 = wave * 16;
    const int m    = lane & 15;
    const int kp   = (lane >> 4) << 1;

    v8f acc = {};
    #pragma unroll 4
    for (int k = 0; k < DH; k += 4) {
        v2f a, b;
        a.x = As[m * DH + k + kp];
        a.y = As[m * DH + k + kp + 1];
        b.x = W[(k + kp)     * DOUT + col0 + m];
        b.y = W[(k + kp + 1) * DOUT + col0 + m];
        acc = __builtin_amdgcn_wmma_f32_16x16x4_f32(false, a, false, b,
                                                    (short)0, acc, false, false);
    }
    const float bv = bias[col0 + m];
    const int rbase = row0 + ((lane >> 4) << 3);
    #pragma unroll
    for (int r = 0; r < 8; ++r)
        OUT[(rbase + r) * DOUT + col0 + m] = acc[r] + bv;
}

// -------------------------------------------------------------- aggregation
// AGG[i,f] = T[i,f] * dinv[i]^2   (self-loop term; also initializes AGG)
__global__ __launch_bounds__(256) void self_init(const float* __restrict__ T,
                                                 const float* __restrict__ dinv,
                                                 float* __restrict__ AGG, int total) {
    int i = blockIdx.x * blockDim.x + threadIdx.x;
    if (i < total) {
        float di = dinv[i >> 7];
        AGG[i] = T[i] * di * di;
    }
}

// one wave per edge: 32 lanes x float4 = 128 features. Gather T[src] (L2-resident),
// scatter-add into AGG[dst] with f32 atomics (L2-resident working set).
__global__ __launch_bounds__(256) void scatter(const int* __restrict__ src,
                                               const int* __restrict__ dst,
                                               const float* __restrict__ ec,
                                               const float* __restrict__ T,
                                               float* __restrict__ AGG, int e) {
    int gid  = blockIdx.x * blockDim.x + threadIdx.x;
    int edge = gid >> 5;
    if (edge >= e) return;
    int lane = gid & 31;
    int s = src[edge], d = dst[edge];
    float c = ec[edge];
    float4 v = ((const float4*)(T + (size_t)s * DH))[lane];   // global_load_b128
    float* a = AGG + (size_t)d * DH + lane * 4;
    atomicAdd(a + 0, v.x * c);
    atomicAdd(a + 1, v.y * c);
    atomicAdd(a + 2, v.z * c);
    atomicAdd(a + 3, v.w * c);
}

// Hout = relu(AGG + b) + Hin
__global__ __launch_bounds__(256) void finalize(const float* __restrict__ AGG,
                                                const float* __restrict__ b,
                                                const float* __restrict__ Hin,
                                                float* __restrict__ Hout, int total) {
    int i = blockIdx.x * blockDim.x + threadIdx.x;
    if (i < total) {
        float v = AGG[i] + b[i & (DH - 1)];
        v = v > 0.0f ? v : 0.0f;
        Hout[i] = v + Hin[i];
    }
}

// ------------------------------------------------------------------ driver
extern "C" void kernel_launch(void* const* d_in, const int* in_sizes, int n_in,
                              void* d_out, int out_size, void* d_ws, size_t ws_size,
                              hipStream_t stream) {
    const float* x    = (const float*)d_in[0];
    const int*   ei   = (const int*)  d_in[1];
    const int*   src  = ei;
    const int*   dst  = ei + NEDGES;
    const float* W0   = (const float*)d_in[2];
    const float* b0   = (const float*)d_in[3];
    const float* W1   = (const float*)d_in[4];
    const float* b1   = (const float*)d_in[5];
    const float* W2   = (const float*)d_in[6];
    const float* b2   = (const float*)d_in[7];
    const float* Wout = (const float*)d_in[8];
    const float* bout = (const float*)d_in[9];

    float* ws   = (float*)d_ws;
    float* deg  = ws;                  ws += NNODES;
    float* dinv = ws;                  ws += NNODES;
    float* ec   = ws;                  ws += NEDGES;
    float* T    = ws;                  ws += (size_t)NNODES * DH;
    float* AGG  = ws;                  ws += (size_t)NNODES * DH;
    float* Ha   = ws;                  ws += (size_t)NNODES * DH;
    float* Hb   = ws;

    const int total   = NNODES * DH;
    const int gElem   = (total + 255) / 256;
    const int gNode   = (NNODES + 255) / 256;
    const int gEdge   = (NEDGES + 255) / 256;
    const int gEdge32 = (int)(((size_t)NEDGES * 32 + 255) / 256);
    const int gTile   = NNODES / 16;   // 50000/16 = 3125, exact

    // degree / normalization coefficients
    zero_f32   <<<gNode,  256, 0, stream>>>(deg, NNODES);
    deg_kernel <<<gEdge,  256, 0, stream>>>(dst, deg, NEDGES);
    dinv_kernel<<<gNode,  256, 0, stream>>>(deg, dinv, NNODES);
    ecoef_kernel<<<gEdge, 256, 0, stream>>>(src, dst, dinv, ec, NEDGES);

    const float* Ws[3] = {W0, W1, W2};
    const float* bs[3] = {b0, b1, b2};
    float*       Ho[3] = {Ha, Hb, Ha};
    const float* hin = x;

    for (int l = 0; l < 3; ++l) {
        gemm_h   <<<gTile,   256, 0, stream>>>(hin, Ws[l], T);
        self_init<<<gElem,   256, 0, stream>>>(T, dinv, AGG, total);
        scatter  <<<gEdge32, 256, 0, stream>>>(src, dst, ec, T, AGG, NEDGES);
        finalize <<<gElem,   256, 0, stream>>>(AGG, bs[l], hin, Ho[l], total);
        hin = Ho[l];
    }

    gemm_o<<<gTile, 128, 0, stream>>>(hin, Wout, bout, (float*)d_out);
}